// Rx_layer_41369124995653
// MI455X (gfx1250) — compile-verified
//
#include <hip/hip_runtime.h>
#include <hip/hip_bf16.h>

// CDNA5 / gfx1250: wave32, WMMA F32 16x16x4 (FP32-exact matrix op).
typedef __attribute__((ext_vector_type(2))) float v2f;
typedef __attribute__((ext_vector_type(8))) float v8f;

#define NQ   10
#define DIM  1024
#define SROW 33          // padded LDS row stride (breaks bank conflicts)

// Accumulate acc += A(32x32) * B(32x32) using V_WMMA_F32_16X16X4_F32.
// A is read row-major from LDS. If BT==true, B is read as Bsrc^T
// (i.e. B[k][n] = Bsrc[n][k]), which is what M * U^T needs.
// Fragment layouts per CDNA5 ISA 7.12.2 (wave32):
//   A 16x4 : lane<16 -> row lane, K = k0+0..1 ; lane>=16 -> row lane-16, K = k0+2..3
//   B 4x16 : lane<16 -> col lane, K = k0+0..1 ; lane>=16 -> col lane-16, K = k0+2..3
//   C/D    : 8 VGPRs, row = 16*mi + (lane>>4)*8 + v, col = 16*ni + (lane&15)
template <bool BT>
__device__ __forceinline__ void mm32_acc(const float* __restrict__ A,
                                         const float* __restrict__ Bsrc,
                                         v8f acc[2][2], int lane)
{
    const int l15 = lane & 15;
    const int kh  = (lane >> 4) << 1;   // 0 or 2
#pragma unroll
    for (int k0 = 0; k0 < 32; k0 += 4) {
        const int kk = k0 + kh;
#pragma unroll
        for (int mi = 0; mi < 2; ++mi) {
            v2f a;
            a.x = A[(mi * 16 + l15) * SROW + kk];
            a.y = A[(mi * 16 + l15) * SROW + kk + 1];
#pragma unroll
            for (int ni = 0; ni < 2; ++ni) {
                v2f b;
                if (BT) {
                    b.x = Bsrc[(ni * 16 + l15) * SROW + kk];
                    b.y = Bsrc[(ni * 16 + l15) * SROW + kk + 1];
                } else {
                    b.x = Bsrc[kk * SROW + ni * 16 + l15];
                    b.y = Bsrc[(kk + 1) * SROW + ni * 16 + l15];
                }
                acc[mi][ni] = __builtin_amdgcn_wmma_f32_16x16x4_f32(
                    false, a, false, b, (short)0, acc[mi][ni], false, false);
            }
        }
    }
}

// One wave per (b,k) problem:
//   out(32x32) = U_head(32x32,cplx) * M(32x32,real) * U_tail^T(32x32,cplx)
// U_head/U_tail entries: (-i)^popcount(i^j) * prod_q (bits equal ? cos : sin)
__global__ void __launch_bounds__(32)
rx_layer_kernel(const float* __restrict__ weights,   // [64][NQ]
                const float* __restrict__ state,     // [64][DIM]
                float* __restrict__ out)             // [64][DIM][2] (re,im)
{
    __shared__ float Mre[32 * SROW];
    __shared__ float UhR[32 * SROW], UhI[32 * SROW];
    __shared__ float UtR[32 * SROW], UtI[32 * SROW];
    __shared__ float TR [32 * SROW], TI [32 * SROW];

    const int lane = threadIdx.x;
    const int p    = blockIdx.x;                 // b*K + k, 0..63
    const float* w   = weights + p * NQ;
    const float* psi = state   + (size_t)p * DIM;
    float*       o   = out     + (size_t)p * DIM * 2;

    // Gate coefficients (lane-uniform).
    float ca[NQ], sb[NQ];
#pragma unroll
    for (int q = 0; q < NQ; ++q) {
        float s, c;
        sincosf(w[q] * 0.5f, &s, &c);
        ca[q] = c;  sb[q] = s;
    }

    // Build U_head, U_tail elementwise and stage the state matrix.
    // Qubit q maps to bit (4-q) of the 5-bit row/col index.
    for (int idx = lane; idx < DIM; idx += 32) {
        const int i = idx >> 5, j = idx & 31, x = i ^ j;
        float ph = 1.0f, pt = 1.0f;
#pragma unroll
        for (int t = 0; t < 5; ++t) {
            const bool d = (x >> t) & 1;
            ph *= d ? sb[4 - t] : ca[4 - t];     // head: qubits 0..4
            pt *= d ? sb[9 - t] : ca[9 - t];     // tail: qubits 5..9
        }
        const int h = __popc(x) & 3;             // phase (-i)^h
        const float pr = (h == 0) ? 1.0f : ((h == 2) ? -1.0f : 0.0f);
        const float pi = (h == 1) ? -1.0f : ((h == 3) ? 1.0f : 0.0f);
        UhR[i * SROW + j] = ph * pr;  UhI[i * SROW + j] = ph * pi;
        UtR[i * SROW + j] = pt * pr;  UtI[i * SROW + j] = pt * pi;
        Mre[i * SROW + j] = psi[idx];
    }
    __syncthreads();

    v8f zero;
#pragma unroll
    for (int e = 0; e < 8; ++e) zero[e] = 0.0f;

    // Phase 1: T = M * U_tail^T  (M real -> only 2 real matmuls)
    v8f accR[2][2], accI[2][2];
#pragma unroll
    for (int mi = 0; mi < 2; ++mi)
#pragma unroll
        for (int ni = 0; ni < 2; ++ni) { accR[mi][ni] = zero; accI[mi][ni] = zero; }

    mm32_acc<true>(Mre, UtR, accR, lane);
    mm32_acc<true>(Mre, UtI, accI, lane);

    const int l15 = lane & 15;
    const int hi8 = (lane >> 4) * 8;
#pragma unroll
    for (int mi = 0; mi < 2; ++mi)
#pragma unroll
        for (int ni = 0; ni < 2; ++ni)
#pragma unroll
            for (int v = 0; v < 8; ++v) {
                const int r = mi * 16 + hi8 + v, c = ni * 16 + l15;
                TR[r * SROW + c] = accR[mi][ni][v];
                TI[r * SROW + c] = accI[mi][ni][v];
            }
    __syncthreads();

    // Phase 2: out = U_head * T (complex*complex).
    // F32 WMMA NEG is C-negate only (ISA 7.12), so use separate accumulators
    // and subtract with VALU instead of neg_a/neg_b.
    v8f P1[2][2], P2[2][2], Q[2][2];
#pragma unroll
    for (int mi = 0; mi < 2; ++mi)
#pragma unroll
        for (int ni = 0; ni < 2; ++ni) { P1[mi][ni] = zero; P2[mi][ni] = zero; Q[mi][ni] = zero; }

    mm32_acc<false>(UhR, TR, P1, lane);   // Re part, +UhR*TR
    mm32_acc<false>(UhI, TI, P2, lane);   // Re part, -UhI*TI (subtract below)
    mm32_acc<false>(UhR, TI, Q,  lane);   // Im part, +UhR*TI
    mm32_acc<false>(UhI, TR, Q,  lane);   // Im part, +UhI*TR

    // Store interleaved complex64 (8B-aligned float2 per amplitude).
#pragma unroll
    for (int mi = 0; mi < 2; ++mi)
#pragma unroll
        for (int ni = 0; ni < 2; ++ni)
#pragma unroll
            for (int v = 0; v < 8; ++v) {
                const int r = mi * 16 + hi8 + v, c = ni * 16 + l15;
                float2 val;
                val.x = P1[mi][ni][v] - P2[mi][ni][v];
                val.y = Q[mi][ni][v];
                reinterpret_cast<float2*>(o)[r * 32 + c] = val;
            }
}

extern "C" void kernel_launch(void* const* d_in, const int* in_sizes, int n_in,
                              void* d_out, int out_size, void* d_ws, size_t ws_size,
                              hipStream_t stream)
{
    const float* weights = (const float*)d_in[0];   // (16,4,1,10,1,1) f32
    const float* state   = (const float*)d_in[1];   // (16,4,1,1024,1) f32
    float* out = (float*)d_out;                     // 64*1024 complex64 -> interleaved f32

    rx_layer_kernel<<<64, 32, 0, stream>>>(weights, state, out);
}